// SModel_74663711473945
// MI455X (gfx1250) — compile-verified
//
#include <hip/hip_runtime.h>

typedef float v8f __attribute__((ext_vector_type(8)));
typedef float v2f __attribute__((ext_vector_type(2)));

// per-node accumulator block (padded to 64 floats = 256B for alignment/L2 locality)
// [0]=count, [1..15]=S1, [16..30]=S2, [31..45]=S3, [46..60]=S4
#define ACC_STRIDE 64

__global__ void zero_ws_kernel(float* __restrict__ w, int n) {
    int i = blockIdx.x * blockDim.x + threadIdx.x;
    if (i < n) w[i] = 0.0f;
}

// ---------------------------------------------------------------------------
// Edge kernel (FULL 16-edge tiles only, no bounds guards).
//   layer1/layer2 via V_WMMA_F32_16X16X4_F32 (K split into 4 chunks of 4).
//   A 16x4 layout (ISA 7.12.2): lane l holds M=l%16, VGPR j / lane-half h -> K=2h+j.
//   B 4x16 layout (mirror):     VGPR j / lane-half h -> row K=2h+j, col N=l%16.
//   C/D 16x16: VGPR r, lane l -> (M = r + 8*(l/16), N = l%16).
// Raw moments scattered with global_atomic_add_f32 into L2-resident accumulators.
// ---------------------------------------------------------------------------
__global__ __launch_bounds__(256) void edge_kernel(
    const float* __restrict__ x_t, const float* __restrict__ edge_attr,
    const int* __restrict__ ei32,                  // low dwords of int64 [2][E]
    const float* __restrict__ w1a, const float* __restrict__ b1a,
    const float* __restrict__ w2a, const float* __restrict__ b2a,
    float* __restrict__ accw, int E, int numFull)
{
    __shared__ float tileBuf[8][16][16];   // per-wave 16 edges x 16 feats
    __shared__ int   srcBuf[8][16];

    const int lane = threadIdx.x & 31;
    const int wave = threadIdx.x >> 5;
    const int gwave = blockIdx.x * (blockDim.x >> 5) + wave;
    const int nwaves = gridDim.x * (blockDim.x >> 5);

    const int col  = lane & 15;   // N (output feature) / edge-local id
    const int half = lane >> 4;   // lane half

    // preload both weight matrices in B layout (4 chunks x 2 VGPRs each)
    float wb1[4][2], wb2[4][2];
#pragma unroll
    for (int c = 0; c < 4; ++c)
#pragma unroll
        for (int j = 0; j < 2; ++j) {
            const int k = 4 * c + 2 * half + j;
            const bool ok = (k < 15) && (col < 15);
            wb1[c][j] = ok ? w1a[k * 15 + col] : 0.0f;
            wb2[c][j] = ok ? w2a[k * 15 + col] : 0.0f;
        }
    const float cb1 = (col < 15) ? b1a[col] : 0.0f;
    const float cb2 = (col < 15) ? b2a[col] : 0.0f;

    for (int t = gwave; t < numFull; t += nwaves) {
        const int base = t * 16;

        // ---- stage concat(x_t[tgt], edge_attr) -> LDS [edge][feat] ----
        {
            const int e  = base + (lane >> 1);
            const int h2 = lane & 1;
            const int tgt = ei32[2 * ((long long)E + e)];
            const float* xr = x_t + (long long)tgt * 5;
            const float* ea = edge_attr + (long long)e * 10;
            float f[8];
            if (h2 == 0) {
                f[0] = xr[0]; f[1] = xr[1]; f[2] = xr[2]; f[3] = xr[3]; f[4] = xr[4];
                f[5] = ea[0]; f[6] = ea[1]; f[7] = ea[2];
            } else {
                f[0] = ea[3]; f[1] = ea[4]; f[2] = ea[5]; f[3] = ea[6];
                f[4] = ea[7]; f[5] = ea[8]; f[6] = ea[9]; f[7] = 0.0f;
            }
#pragma unroll
            for (int i = 0; i < 8; ++i) tileBuf[wave][lane >> 1][h2 * 8 + i] = f[i];
            if (lane < 16) srcBuf[wave][lane] = ei32[2 * (base + lane)];
        }
        // wave-private LDS region; DS ops are in-order per wave -> no barrier

        // ---- A chunks for layer 1 ----
        v2f a[4];
#pragma unroll
        for (int c = 0; c < 4; ++c) {
            a[c].x = tileBuf[wave][col][4 * c + 2 * half + 0];
            a[c].y = tileBuf[wave][col][4 * c + 2 * half + 1];
        }

        // ---- layer 1: D = sum_c A_c x B_c + bias ----
        v8f acc;
#pragma unroll
        for (int i = 0; i < 8; ++i) acc[i] = cb1;
#pragma unroll
        for (int c = 0; c < 4; ++c) {
            v2f b; b.x = wb1[c][0]; b.y = wb1[c][1];
            acc = __builtin_amdgcn_wmma_f32_16x16x4_f32(
                false, a[c], false, b, (short)0, acc, false, false);
        }
        // LeakyReLU(0.1) == max(x, 0.1x)
#pragma unroll
        for (int i = 0; i < 8; ++i) acc[i] = fmaxf(acc[i], 0.1f * acc[i]);

        // ---- repack D (edges x feats) -> A layout via LDS ----
#pragma unroll
        for (int r = 0; r < 8; ++r) tileBuf[wave][r + 8 * half][col] = acc[r];
#pragma unroll
        for (int c = 0; c < 4; ++c) {
            a[c].x = tileBuf[wave][col][4 * c + 2 * half + 0];
            a[c].y = tileBuf[wave][col][4 * c + 2 * half + 1];
        }

        // ---- layer 2 ----
#pragma unroll
        for (int i = 0; i < 8; ++i) acc[i] = cb2;
#pragma unroll
        for (int c = 0; c < 4; ++c) {
            v2f b; b.x = wb2[c][0]; b.y = wb2[c][1];
            acc = __builtin_amdgcn_wmma_f32_16x16x4_f32(
                false, a[c], false, b, (short)0, acc, false, false);
        }

        // ---- bulk-read the 8 src indices this lane scatters to ----
        int s[8];
        {
            const int4 sA = *(const int4*)&srcBuf[wave][8 * half];
            const int4 sB = *(const int4*)&srcBuf[wave][8 * half + 4];
            s[0] = sA.x; s[1] = sA.y; s[2] = sA.z; s[3] = sA.w;
            s[4] = sB.x; s[5] = sB.y; s[6] = sB.z; s[7] = sB.w;
        }

        // ---- scatter: feature lanes add moments, the col==15 pad lanes add counts ----
        if (col < 15) {
#pragma unroll
            for (int r = 0; r < 8; ++r) {
                const float m  = acc[r];
                const float m2 = m * m;
                float* p = accw + (long long)s[r] * ACC_STRIDE;
                atomicAdd(p + 1  + col, m);
                atomicAdd(p + 16 + col, m2);
                atomicAdd(p + 31 + col, m2 * m);
                atomicAdd(p + 46 + col, m2 * m2);
            }
        } else {
#pragma unroll
            for (int r = 0; r < 8; ++r)
                atomicAdd(accw + (long long)s[r] * ACC_STRIDE, 1.0f);
        }
    }
}

// ---------------------------------------------------------------------------
// Scalar tail kernel for E % 16 leftover edges (launched only if needed).
// ---------------------------------------------------------------------------
__global__ void edge_tail_kernel(
    const float* __restrict__ x_t, const float* __restrict__ edge_attr,
    const int* __restrict__ ei32,
    const float* __restrict__ w1a, const float* __restrict__ b1a,
    const float* __restrict__ w2a, const float* __restrict__ b2a,
    float* __restrict__ accw, int E, int tailStart)
{
    const int e = tailStart + blockIdx.x * blockDim.x + threadIdx.x;
    if (e >= E) return;
    const int src = ei32[2 * e];
    const int tgt = ei32[2 * ((long long)E + e)];

    float in[15];
#pragma unroll
    for (int i = 0; i < 5; ++i) in[i] = x_t[(long long)tgt * 5 + i];
#pragma unroll
    for (int i = 0; i < 10; ++i) in[5 + i] = edge_attr[(long long)e * 10 + i];

    float h1[15];
#pragma unroll
    for (int j = 0; j < 15; ++j) {
        float z = b1a[j];
        for (int k = 0; k < 15; ++k) z = fmaf(in[k], w1a[k * 15 + j], z);
        h1[j] = fmaxf(z, 0.1f * z);
    }
    float* p = accw + (long long)src * ACC_STRIDE;
    atomicAdd(p, 1.0f);
#pragma unroll
    for (int j = 0; j < 15; ++j) {
        float m = b2a[j];
        for (int k = 0; k < 15; ++k) m = fmaf(h1[k], w2a[k * 15 + j], m);
        const float m2 = m * m;
        atomicAdd(p + 1  + j, m);
        atomicAdd(p + 16 + j, m2);
        atomicAdd(p + 31 + j, m2 * m);
        atomicAdd(p + 46 + j, m2 * m2);
    }
}

// ---------------------------------------------------------------------------
// Node kernel: raw moments -> (count, mean, std, skew, kurt); 81->10->10 MLP.
// ---------------------------------------------------------------------------
__global__ __launch_bounds__(256) void node_kernel(
    const float* __restrict__ x_s, const float* __restrict__ u,
    const long long* __restrict__ batch_s,
    const float* __restrict__ w1b, const float* __restrict__ b1b,
    const float* __restrict__ w2b, const float* __restrict__ b2b,
    const float* __restrict__ accw, float* __restrict__ out, int n)
{
    __shared__ float sw1[81 * 10];
    __shared__ float sb1[10];
    __shared__ float sw2[100];
    __shared__ float sb2[10];
    for (int i = threadIdx.x; i < 810; i += blockDim.x) sw1[i] = w1b[i];
    for (int i = threadIdx.x; i < 100; i += blockDim.x) sw2[i] = w2b[i];
    if (threadIdx.x < 10) { sb1[threadIdx.x] = b1b[threadIdx.x]; sb2[threadIdx.x] = b2b[threadIdx.x]; }
    __syncthreads();

    const int node = blockIdx.x * blockDim.x + threadIdx.x;
    if (node >= n) return;

    const float* p = accw + (long long)node * ACC_STRIDE;
    const float cnt = p[0];
    const float inv = 1.0f / fmaxf(cnt, 1.0f);

    float h[81];
    const float* xs = x_s + (long long)node * 10;
#pragma unroll
    for (int i = 0; i < 10; ++i) h[i] = xs[i];
    h[10] = cnt;

#pragma unroll
    for (int f = 0; f < 15; ++f) {
        const float s1 = p[1 + f], s2 = p[16 + f], s3 = p[31 + f], s4 = p[46 + f];
        const float mean  = s1 * inv;
        const float mean2 = s2 * inv;
        const float var   = fmaxf(mean2 - mean * mean, 0.0f);
        const float stdv  = sqrtf(var + 1e-6f);
        const float mu2   = mean * mean;
        const float m3 = (s3 - 3.0f * mean * s2) * inv + 2.0f * mu2 * mean;
        const float m4 = (s4 - 4.0f * mean * s3 + 6.0f * mu2 * s2) * inv - 3.0f * mu2 * mu2;
        const float is  = 1.0f / stdv;
        const float is2 = is * is;
        h[11 + f] = mean;
        h[26 + f] = stdv;
        h[41 + f] = m3 * is2 * is;
        h[56 + f] = m4 * is2 * is2;
    }
    const long long b = batch_s[node];
    const float* ur = u + b * 10;
#pragma unroll
    for (int i = 0; i < 10; ++i) h[71 + i] = ur[i];

    float z[10];
#pragma unroll
    for (int j = 0; j < 10; ++j) z[j] = sb1[j];
    for (int k = 0; k < 81; ++k) {
        const float hk = h[k];
#pragma unroll
        for (int j = 0; j < 10; ++j) z[j] = fmaf(hk, sw1[k * 10 + j], z[j]);
    }
#pragma unroll
    for (int j = 0; j < 10; ++j) z[j] = fmaxf(z[j], 0.1f * z[j]);

    float o[10];
#pragma unroll
    for (int j = 0; j < 10; ++j) o[j] = sb2[j];
#pragma unroll
    for (int k = 0; k < 10; ++k) {
        const float zk = z[k];
#pragma unroll
        for (int j = 0; j < 10; ++j) o[j] = fmaf(zk, sw2[k * 10 + j], o[j]);
    }
    float* op = out + (long long)node * 10;
#pragma unroll
    for (int j = 0; j < 10; ++j) op[j] = o[j];
}

extern "C" void kernel_launch(void* const* d_in, const int* in_sizes, int n_in,
                              void* d_out, int out_size, void* d_ws, size_t ws_size,
                              hipStream_t stream) {
    const float*     x_s       = (const float*)d_in[0];
    const float*     x_t       = (const float*)d_in[1];
    const float*     edge_attr = (const float*)d_in[2];
    const float*     u         = (const float*)d_in[3];
    const int*       ei32      = (const int*)d_in[4];       // int64 viewed as dword pairs
    const long long* batch_s   = (const long long*)d_in[5];
    const float*     w1a = (const float*)d_in[6];
    const float*     b1a = (const float*)d_in[7];
    const float*     w2a = (const float*)d_in[8];
    const float*     b2a = (const float*)d_in[9];
    const float*     w1b = (const float*)d_in[10];
    const float*     b1b = (const float*)d_in[11];
    const float*     w2b = (const float*)d_in[12];
    const float*     b2b = (const float*)d_in[13];

    const int nS = in_sizes[0] / 10;   // 100000
    const int E  = in_sizes[4] / 2;    // 4,000,000

    float* accw = (float*)d_ws;        // nS * 64 floats = 25.6 MB (fits L2)
    const int accN = nS * ACC_STRIDE;
    zero_ws_kernel<<<(accN + 255) / 256, 256, 0, stream>>>(accw, accN);

    const int numFull = E / 16;
    edge_kernel<<<2048, 256, 0, stream>>>(x_t, edge_attr, ei32,
                                          w1a, b1a, w2a, b2a, accw, E, numFull);
    const int rem = E - numFull * 16;
    if (rem > 0) {
        edge_tail_kernel<<<1, 32, 0, stream>>>(x_t, edge_attr, ei32,
                                               w1a, b1a, w2a, b2a, accw,
                                               E, numFull * 16);
    }

    node_kernel<<<(nS + 255) / 256, 256, 0, stream>>>(x_s, u, batch_s,
                                                      w1b, b1b, w2b, b2b,
                                                      accw, (float*)d_out, nS);
}